// Model_38620345925653
// MI455X (gfx1250) — compile-verified
//
#include <hip/hip_runtime.h>
#include <hip/hip_bf16.h>
#include <math.h>

// Model dims
constexpr int B = 16, T = 512, N = 128, PRED = 256, L = 768, DM = 32;
constexpr float EPS = 1e-5f;

typedef float v2f __attribute__((ext_vector_type(2)));
typedef float v8f __attribute__((ext_vector_type(8)));

// ---------------------------------------------------------------------------
// K1: per-(b,n) mean / stdev over T
// ---------------------------------------------------------------------------
__global__ __launch_bounds__(128) void k_stats(const float* __restrict__ bx,
                                               float* __restrict__ mean,
                                               float* __restrict__ stdev) {
    int b = blockIdx.x;
    int n = threadIdx.x;
    const float* p = bx + (size_t)b * T * N + n;
    float s = 0.f, s2 = 0.f;
    for (int t = 0; t < T; ++t) {
        float v = p[(size_t)t * N];
        s += v; s2 += v * v;
    }
    float mu  = s / (float)T;
    float var = s2 / (float)T - mu * mu;
    mean[b * N + n]  = mu;
    stdev[b * N + n] = sqrtf(var + EPS);
}

// ---------------------------------------------------------------------------
// K0: simple transpose  src(R,C) -> dst(C,R)
// ---------------------------------------------------------------------------
__global__ __launch_bounds__(256) void k_transpose(const float* __restrict__ src,
                                                   float* __restrict__ dst,
                                                   int R, int C) {
    int idx = blockIdx.x * blockDim.x + threadIdx.x;
    if (idx < R * C) {
        int r = idx / C, c = idx % C;
        dst[(size_t)c * R + r] = src[idx];
    }
}

// ---------------------------------------------------------------------------
// K2: GEMM1 via V_WMMA_F32_16X16X4_F32, 1x4 N-blocking per wave.
//   x[(b,n), l] = sum_t ((bx[b,t,n]-mu)/sd) * leT[t,l] + le_b[l]
//   M = B*N = 2048, Ncols = L = 768, K = T = 512
//   Each wave computes a 16(M) x 64(N) tile: one A-fragment load feeds 4 WMMAs.
// ---------------------------------------------------------------------------
__global__ __launch_bounds__(256) void k_gemm1(const float* __restrict__ bx,
                                               const float* __restrict__ mean,
                                               const float* __restrict__ stdev,
                                               const float* __restrict__ leT,   // (T, L)
                                               const float* __restrict__ le_b,  // (L)
                                               float* __restrict__ x) {         // (B*N, L)
    const int groupsN = L / 64;                      // 12 groups of 4 tiles
    int wave = blockIdx.x * (blockDim.x >> 5) + (threadIdx.x >> 5);
    int lane = threadIdx.x & 31;
    int tm = wave / groupsN;
    int tg = wave % groupsN;
    if (tm >= (B * N) / 16) return;

    int row = tm * 16 + (lane & 15);                 // A-frag row (M)
    int b   = row >> 7;
    int n   = row & 127;
    float mu  = mean[row];
    float inv = 1.0f / stdev[row];
    int kOff  = (lane >= 16) ? 2 : 0;
    int col0  = tg * 64 + (lane & 15);               // first of 4 cols (stride 16)

    const float* ap = bx + (size_t)b * T * N + n;
    v8f acc0 = {}, acc1 = {}, acc2 = {}, acc3 = {};
    for (int k = 0; k < T; k += 4) {
        int t0 = k + kOff;
        v2f a;
        a.x = (ap[(size_t)t0 * N] - mu) * inv;
        a.y = (ap[(size_t)(t0 + 1) * N] - mu) * inv;
        const float* bp0 = leT + (size_t)t0 * L + col0;
        const float* bp1 = leT + (size_t)(t0 + 1) * L + col0;
        v2f b0, b1, b2, b3;
        b0.x = bp0[0];  b0.y = bp1[0];
        b1.x = bp0[16]; b1.y = bp1[16];
        b2.x = bp0[32]; b2.y = bp1[32];
        b3.x = bp0[48]; b3.y = bp1[48];
        acc0 = __builtin_amdgcn_wmma_f32_16x16x4_f32(false, a, false, b0, (short)0, acc0, false, false);
        acc1 = __builtin_amdgcn_wmma_f32_16x16x4_f32(false, a, false, b1, (short)0, acc1, false, false);
        acc2 = __builtin_amdgcn_wmma_f32_16x16x4_f32(false, a, false, b2, (short)0, acc2, false, false);
        acc3 = __builtin_amdgcn_wmma_f32_16x16x4_f32(false, a, false, b3, (short)0, acc3, false, false);
    }
    v8f accs[4] = { acc0, acc1, acc2, acc3 };
#pragma unroll
    for (int j = 0; j < 4; ++j) {
        int col = col0 + j * 16;
        float bias = le_b[col];
#pragma unroll
        for (int r = 0; r < 8; ++r) {
            int m = tm * 16 + r + ((lane >= 16) ? 8 : 0);
            x[(size_t)m * L + col] = accs[j][r] + bias;
        }
    }
}

// ---------------------------------------------------------------------------
// K3/K7: strided conv1d (stride == win, VALID) : out[b,o,j] = b[o] + sum_{i,k}
// ---------------------------------------------------------------------------
__global__ __launch_bounds__(128) void k_conv1d(const float* __restrict__ xin,  // (B,N,inLen)
                                                const float* __restrict__ w,    // (N,N,win)
                                                const float* __restrict__ bias,
                                                float* __restrict__ out,        // (B,N,outLen)
                                                int win, int outLen) {
    __shared__ float xs[128 * 12];
    int bj = blockIdx.x;
    int b  = bj / outLen;
    int j  = bj % outLen;
    int o  = threadIdx.x;
    int inLen = outLen * win;
    for (int idx = threadIdx.x; idx < N * win; idx += blockDim.x) {
        int i = idx / win, k = idx % win;
        xs[idx] = xin[(size_t)b * N * inLen + (size_t)i * inLen + j * win + k];
    }
    __syncthreads();
    float acc = bias[o];
    const float* wp = w + (size_t)o * N * win;
    int total = N * win;
    for (int ik = 0; ik < total; ++ik) acc += xs[ik] * wp[ik];
    out[(size_t)b * N * outLen + (size_t)o * outLen + j] = acc;
}

// ---------------------------------------------------------------------------
// K4/K8: fused BatchNorm1d (stats over (B, len)) + ELU, in place. 1 block/chan
// ---------------------------------------------------------------------------
__global__ __launch_bounds__(256) void k_bn_elu(float* __restrict__ t,          // (B,N,len)
                                                const float* __restrict__ g,
                                                const float* __restrict__ beta,
                                                int len) {
    __shared__ float sh[256], sh2[256];
    int c = blockIdx.x, tid = threadIdx.x;
    int total = B * len;
    float s = 0.f, s2 = 0.f;
    for (int idx = tid; idx < total; idx += blockDim.x) {
        int b = idx / len, j = idx % len;
        float v = t[(size_t)b * N * len + (size_t)c * len + j];
        s += v; s2 += v * v;
    }
    sh[tid] = s; sh2[tid] = s2;
    __syncthreads();
    for (int off = 128; off > 0; off >>= 1) {
        if (tid < off) { sh[tid] += sh[tid + off]; sh2[tid] += sh2[tid + off]; }
        __syncthreads();
    }
    float mu  = sh[0] / (float)total;
    float var = sh2[0] / (float)total - mu * mu;
    float scl = rsqrtf(var + EPS) * g[c];
    float shf = beta[c];
    for (int idx = tid; idx < total; idx += blockDim.x) {
        int b = idx / len, j = idx % len;
        size_t off = (size_t)b * N * len + (size_t)c * len + j;
        float v = (t[off] - mu) * scl + shf;
        t[off] = (v > 0.f) ? v : (expf(v) - 1.0f);    // ELU(alpha=1)
    }
}

// ---------------------------------------------------------------------------
// K5/K9: cubic-spline second derivatives.  A = 4I + offdiag(1) tridiagonal.
//   One block per batch b; thread = channel c. Thomas pivots are constant.
//   M out: (B, Lk, N), zero-padded at j=0 and j=Lk-1.
// ---------------------------------------------------------------------------
__global__ __launch_bounds__(128) void k_spline_solve(const float* __restrict__ t, // (B,N,Lk)
                                                      float* __restrict__ M,       // (B,Lk,N)
                                                      int Lk) {
    __shared__ float d[128 * 62];
    __shared__ float cps[64];
    int b = blockIdx.x;
    int c = threadIdx.x;
    int m = Lk - 2;
    float r6h2 = 6.0f * (float)(Lk - 1) * (float)(Lk - 1);   // 6/h^2
    if (c == 0) {
        float p = 0.f;
        for (int i = 0; i < m; ++i) { p = 1.0f / (4.0f - p); cps[i] = p; }
    }
    __syncthreads();
    const float* y = t + (size_t)b * N * Lk + (size_t)c * Lk;
    float* dc = d + c * 62;
    float dprev = 0.f;
    for (int i = 0; i < m; ++i) {
        float rhs = (y[i + 2] - 2.0f * y[i + 1] + y[i]) * r6h2;
        float denom = (i == 0) ? 4.0f : (4.0f - cps[i - 1]);
        float dv = (rhs - dprev) / denom;
        dc[i] = dv; dprev = dv;
    }
    float Mn = 0.f;
    for (int i = m - 1; i >= 0; --i) {
        float Mi = dc[i] - cps[i] * Mn;
        M[(size_t)b * Lk * N + (size_t)(i + 1) * N + c] = Mi;
        Mn = Mi;
    }
    M[(size_t)b * Lk * N + c] = 0.f;
    M[(size_t)b * Lk * N + (size_t)(Lk - 1) * N + c] = 0.f;
}

// ---------------------------------------------------------------------------
// K6/K10: spline evaluation to L points -> (B,N,L)
// ---------------------------------------------------------------------------
__global__ __launch_bounds__(256) void k_spline_eval(const float* __restrict__ t, // (B,N,Lk)
                                                     const float* __restrict__ M, // (B,Lk,N)
                                                     float* __restrict__ s,       // (B,N,L)
                                                     int Lk) {
    int gid = blockIdx.x * blockDim.x + threadIdx.x;
    if (gid >= B * N * L) return;
    int q  = gid % L;
    int bc = gid / L;
    int c  = bc % N;
    int b  = bc / N;
    float h  = 1.0f / (float)(Lk - 1);
    float qq = (float)q / (float)(L - 1);
    int idx = (int)floorf(qq * (float)(Lk - 1));
    idx = idx < 0 ? 0 : (idx > Lk - 2 ? Lk - 2 : idx);
    float ss = qq - (float)idx * h;
    float u  = h - ss;
    const float* y = t + (size_t)b * N * Lk + (size_t)c * Lk;
    float yi  = y[idx], yi1 = y[idx + 1];
    float M0  = M[(size_t)b * Lk * N + (size_t)idx * N + c];
    float M1v = M[(size_t)b * Lk * N + (size_t)(idx + 1) * N + c];
    float hh6 = h * h / 6.0f;
    float val = (M0 * u * u * u + M1v * ss * ss * ss) / (6.0f * h)
              + (yi - M0 * hh6) * (u / h) + (yi1 - M1v * hh6) * (ss / h);
    s[(size_t)bc * L + q] = val;
}

// ---------------------------------------------------------------------------
// K11: fused 2-layer 2D conv stack.  a = [x, s1, s2] viewed as (B,3,L,N).
//   h1 = gelu(0.5*(1x1 + 3x3))  (32 ch, kept entirely in LDS — never to HBM)
//   h2 = 0.5*(1x1 + 3x3) over h1 ; co = x + h2
// ---------------------------------------------------------------------------
__global__ __launch_bounds__(256) void k_conv2d_fused(
        const float* __restrict__ x, const float* __restrict__ s1,
        const float* __restrict__ s2,
        const float* __restrict__ i1w0, const float* __restrict__ i1b0,
        const float* __restrict__ i1w1, const float* __restrict__ i1b1,
        const float* __restrict__ i2w0, const float* __restrict__ i2b0,
        const float* __restrict__ i2w1, const float* __restrict__ i2b1,
        float* __restrict__ co) {
    // layouts: aT[c][ww][yy] (yy contiguous), h1T[dm][ww][yy]
    __shared__ float aT[3][20][20];
    __shared__ float h1T[32][18][18];
    __shared__ float w0s[32 * 3], w1s[32 * 27], b0s[32], b1s[32];
    __shared__ float w20s[32], w21s[32 * 9];

    int b  = blockIdx.z;
    int Y0 = blockIdx.y * 16;          // along L (H)
    int W0 = blockIdx.x * 16;          // along N (W)
    int tid = threadIdx.x;

    for (int i = tid; i < 96;  i += 256) w0s[i] = i1w0[i];
    for (int i = tid; i < 864; i += 256) w1s[i] = i1w1[i];
    for (int i = tid; i < 288; i += 256) w21s[i] = i2w1[i];
    if (tid < 32) { b0s[tid] = i1b0[tid]; b1s[tid] = i1b1[tid]; w20s[tid] = i2w0[tid]; }

    const float* chans[3] = { x, s1, s2 };
    for (int idx = tid; idx < 3 * 20 * 20; idx += 256) {
        int c   = idx / 400;
        int rem = idx % 400;
        int ww  = rem / 20;
        int yy  = rem % 20;
        int gy = Y0 - 2 + yy;
        int gw = W0 - 2 + ww;
        float v = 0.f;
        if (gy >= 0 && gy < L && gw >= 0 && gw < N)
            v = chans[c][(size_t)b * N * L + (size_t)gw * L + gy];
        aT[c][ww][yy] = v;
    }
    __syncthreads();

    for (int idx = tid; idx < 32 * 18 * 18; idx += 256) {
        int dm  = idx / 324;
        int rem = idx % 324;
        int ww  = rem / 18;
        int yy  = rem % 18;
        int gy = Y0 - 1 + yy;
        int gw = W0 - 1 + ww;
        float v = 0.f;
        if (gy >= 0 && gy < L && gw >= 0 && gw < N) {
            float a0 = b0s[dm];
            float a1 = b1s[dm];
#pragma unroll
            for (int c = 0; c < 3; ++c) {
                a0 += w0s[dm * 3 + c] * aT[c][ww + 1][yy + 1];
#pragma unroll
                for (int ky = 0; ky < 3; ++ky)
#pragma unroll
                    for (int kx = 0; kx < 3; ++kx)
                        a1 += w1s[((dm * 3 + c) * 3 + ky) * 3 + kx] * aT[c][ww + kx][yy + ky];
            }
            float hv = 0.5f * (a0 + a1);
            v = 0.5f * hv * (1.0f + erff(hv * 0.70710678118654752f));   // exact GELU
        }
        h1T[dm][ww][yy] = v;
    }
    __syncthreads();

    {
        int ww = tid / 16;
        int yy = tid % 16;
        float a0 = i2b0[0];
        float a1 = i2b1[0];
        for (int dm = 0; dm < 32; ++dm) {
            a0 += w20s[dm] * h1T[dm][ww + 1][yy + 1];
#pragma unroll
            for (int ky = 0; ky < 3; ++ky)
#pragma unroll
                for (int kx = 0; kx < 3; ++kx)
                    a1 += w21s[dm * 9 + ky * 3 + kx] * h1T[dm][ww + kx][yy + ky];
        }
        float h2v = 0.5f * (a0 + a1);
        int gy = Y0 + yy, gw = W0 + ww;
        size_t off = (size_t)b * N * L + (size_t)gw * L + gy;
        co[off] = x[off] + h2v;
    }
}

// ---------------------------------------------------------------------------
// K12: GEMM2 via V_WMMA_F32_16X16X4_F32, 1x4 N-blocking + de-normalization
//   out[b,p,n] = (sum_l co[(b,n),l]*ldT[l,p] + ld_b[p]) * sd[b,n] + mu[b,n]
// ---------------------------------------------------------------------------
__global__ __launch_bounds__(256) void k_gemm2(const float* __restrict__ co,   // (B*N, L)
                                               const float* __restrict__ ldT,  // (L, PRED)
                                               const float* __restrict__ ld_b,
                                               const float* __restrict__ mean,
                                               const float* __restrict__ stdev,
                                               float* __restrict__ out) {      // (B,PRED,N)
    const int groupsN = PRED / 64;                   // 4 groups of 4 tiles
    int wave = blockIdx.x * (blockDim.x >> 5) + (threadIdx.x >> 5);
    int lane = threadIdx.x & 31;
    int tm = wave / groupsN;
    int tg = wave % groupsN;
    if (tm >= (B * N) / 16) return;

    int row  = tm * 16 + (lane & 15);
    int kOff = (lane >= 16) ? 2 : 0;
    int col0 = tg * 64 + (lane & 15);

    const float* ap = co + (size_t)row * L;
    v8f acc0 = {}, acc1 = {}, acc2 = {}, acc3 = {};
    for (int k = 0; k < L; k += 4) {
        int t0 = k + kOff;
        v2f a;
        a.x = ap[t0];
        a.y = ap[t0 + 1];
        const float* bp0 = ldT + (size_t)t0 * PRED + col0;
        const float* bp1 = ldT + (size_t)(t0 + 1) * PRED + col0;
        v2f b0, b1, b2, b3;
        b0.x = bp0[0];  b0.y = bp1[0];
        b1.x = bp0[16]; b1.y = bp1[16];
        b2.x = bp0[32]; b2.y = bp1[32];
        b3.x = bp0[48]; b3.y = bp1[48];
        acc0 = __builtin_amdgcn_wmma_f32_16x16x4_f32(false, a, false, b0, (short)0, acc0, false, false);
        acc1 = __builtin_amdgcn_wmma_f32_16x16x4_f32(false, a, false, b1, (short)0, acc1, false, false);
        acc2 = __builtin_amdgcn_wmma_f32_16x16x4_f32(false, a, false, b2, (short)0, acc2, false, false);
        acc3 = __builtin_amdgcn_wmma_f32_16x16x4_f32(false, a, false, b3, (short)0, acc3, false, false);
    }
    v8f accs[4] = { acc0, acc1, acc2, acc3 };
#pragma unroll
    for (int j = 0; j < 4; ++j) {
        int col = col0 + j * 16;
        float bias = ld_b[col];
#pragma unroll
        for (int r = 0; r < 8; ++r) {
            int m = tm * 16 + r + ((lane >= 16) ? 8 : 0);
            int bb = m >> 7, n = m & 127;
            float v = (accs[j][r] + bias) * stdev[m] + mean[m];
            out[(size_t)bb * PRED * N + (size_t)col * N + n] = v;
        }
    }
}

// ---------------------------------------------------------------------------
extern "C" void kernel_launch(void* const* d_in, const int* in_sizes, int n_in,
                              void* d_out, int out_size, void* d_ws, size_t ws_size,
                              hipStream_t stream) {
    const float* bx    = (const float*)d_in[0];
    const float* le_w  = (const float*)d_in[4];
    const float* le_b  = (const float*)d_in[5];
    const float* c1_w  = (const float*)d_in[6];
    const float* c1_b  = (const float*)d_in[7];
    const float* bn1_g = (const float*)d_in[8];
    const float* bn1_b = (const float*)d_in[9];
    const float* c2_w  = (const float*)d_in[10];
    const float* c2_b  = (const float*)d_in[11];
    const float* bn2_g = (const float*)d_in[12];
    const float* bn2_b = (const float*)d_in[13];
    const float* i1w0  = (const float*)d_in[14];
    const float* i1b0  = (const float*)d_in[15];
    const float* i1w1  = (const float*)d_in[16];
    const float* i1b1  = (const float*)d_in[17];
    const float* i2w0  = (const float*)d_in[18];
    const float* i2b0  = (const float*)d_in[19];
    const float* i2w1  = (const float*)d_in[20];
    const float* i2b1  = (const float*)d_in[21];
    const float* ld_w  = (const float*)d_in[22];
    const float* ld_b  = (const float*)d_in[23];
    float* out = (float*)d_out;

    float* ws    = (float*)d_ws;
    float* mean  = ws;                      // 2048
    float* stdev = mean  + 2048;            // 2048
    float* leT   = stdev + 2048;            // T*L    = 393216
    float* ldT   = leT   + (size_t)T * L;   // L*PRED = 196608
    float* x     = ldT   + (size_t)L * PRED;// B*N*L  = 1572864
    float* t1    = x     + (size_t)B * N * L;   // B*N*64 = 131072
    float* t2    = t1    + (size_t)B * N * 64;  // B*N*16 = 32768
    float* M1    = t2    + (size_t)B * N * 16;  // B*64*N = 131072
    float* M2    = M1    + (size_t)B * 64 * N;  // B*16*N = 32768
    float* s1    = M2    + (size_t)B * 16 * N;  // B*N*L
    float* s2    = s1    + (size_t)B * N * L;   // B*N*L
    float* co    = s2    + (size_t)B * N * L;   // B*N*L

    // 1. normalization stats
    k_stats<<<B, 128, 0, stream>>>(bx, mean, stdev);
    // 2. transpose weights for coalesced WMMA B-fragments
    k_transpose<<<(L * T + 255) / 256, 256, 0, stream>>>(le_w, leT, L, T);
    k_transpose<<<(PRED * L + 255) / 256, 256, 0, stream>>>(ld_w, ldT, PRED, L);
    // 3. GEMM1 (WMMA f32): 128 x 12 wave-tiles (16x64 each), 8 waves/block
    k_gemm1<<<(128 * 12) / 8, 256, 0, stream>>>(bx, mean, stdev, leT, le_b, x);
    // 4. scale pyramid 1: conv(k=12,s=12) -> BN+ELU -> spline to L
    k_conv1d<<<B * 64, 128, 0, stream>>>(x, c1_w, c1_b, t1, 12, 64);
    k_bn_elu<<<N, 256, 0, stream>>>(t1, bn1_g, bn1_b, 64);
    k_spline_solve<<<B, 128, 0, stream>>>(t1, M1, 64);
    k_spline_eval<<<(B * N * L + 255) / 256, 256, 0, stream>>>(t1, M1, s1, 64);
    // 5. scale pyramid 2: conv(k=4,s=4) -> BN+ELU -> spline to L
    k_conv1d<<<B * 16, 128, 0, stream>>>(t1, c2_w, c2_b, t2, 4, 16);
    k_bn_elu<<<N, 256, 0, stream>>>(t2, bn2_g, bn2_b, 16);
    k_spline_solve<<<B, 128, 0, stream>>>(t2, M2, 16);
    k_spline_eval<<<(B * N * L + 255) / 256, 256, 0, stream>>>(t2, M2, s2, 16);
    // 6. fused 2D conv stack (h1 lives only in LDS), residual add into co
    dim3 g(N / 16, L / 16, B);
    k_conv2d_fused<<<g, 256, 0, stream>>>(x, s1, s2, i1w0, i1b0, i1w1, i1b1,
                                          i2w0, i2b0, i2w1, i2b1, co);
    // 7. GEMM2 (WMMA f32, 16x64 per wave) + de-normalization
    k_gemm2<<<(128 * 4) / 8, 256, 0, stream>>>(co, ldT, ld_b, mean, stdev, out);
}